// LocalDists_68247030334257
// MI455X (gfx1250) — compile-verified
//
#include <hip/hip_runtime.h>
#include <hip/hip_bf16.h>

typedef float v2f __attribute__((ext_vector_type(2)));
typedef float v8f __attribute__((ext_vector_type(8)));

#define NPTS 8192
#define KNB  16            // neighbors per side -> 2*KNB = 32 per point
#define PTS_PER_BLOCK 128  // 8 wave32 tiles of 16 points

// One block: 256 threads = 8 waves; each wave owns a 16-point tile.
// Per wave: 3x V_WMMA_F32_16X16X4_F32 computes the 16x48 Gram band
// (neighbor columns tileStart-16 .. tileStart+31). d2 via stored |x|^2,
// then per-point 32-element bitonic sort in registers.
__global__ __launch_bounds__(256) void LocalDists_band_wmma(
    const float* __restrict__ x,   // [B, NPTS, 3]
    float* __restrict__ out)       // [B, NPTS*32]
{
    const int blocksPerBatch = NPTS / PTS_PER_BLOCK;        // 64
    const int b          = blockIdx.x / blocksPerBatch;
    const int blockStart = (blockIdx.x % blocksPerBatch) * PTS_PER_BLOCK;
    const float* xb = x + (size_t)b * NPTS * 3;

    __shared__ float sq[PTS_PER_BLOCK + 32];                // |x|^2 for [blockStart-16, blockStart+144)
    __shared__ float drow[8][16][48];                       // per wave: 16 rows x 48 neighbor cols

    const int tid = threadIdx.x;

    // ---- squared-norm table for the whole block's neighbor range ----
    if (tid < PTS_PER_BLOCK + 32) {
        int g  = blockStart - 16 + tid;
        int gc = g < 0 ? 0 : (g > NPTS - 1 ? NPTS - 1 : g);
        float a0 = xb[gc * 3 + 0];
        float a1 = xb[gc * 3 + 1];
        float a2 = xb[gc * 3 + 2];
        sq[tid] = a0 * a0 + a1 * a1 + a2 * a2;
    }
    __syncthreads();

    const int wave = tid >> 5;
    const int lane = tid & 31;
    const int n    = lane & 15;   // row (for A) / col (for B) index inside tile
    const int hi   = lane >> 4;   // K-half select per ISA 16x4 f32 layout
    const int tileStart = blockStart + wave * 16;

    // ---- A matrix: 16x4 f32 (rows = tile points, K = {x,y,z,0}) ----
    // lanes 0-15: VGPR0=K0, VGPR1=K1 ; lanes 16-31: VGPR0=K2, VGPR1=K3(pad)
    v2f a;
    {
        int i = tileStart + n;
        if (hi == 0) { a.x = xb[i * 3 + 0]; a.y = xb[i * 3 + 1]; }
        else         { a.x = xb[i * 3 + 2]; a.y = 0.0f;          }
    }

    // ---- 3 neighbor tiles -> 3 WMMAs, convert gram -> distance, stash in LDS ----
    #pragma unroll
    for (int tIdx = 0; tIdx < 3; ++tIdx) {
        int jt = tileStart - 16 + tIdx * 16;   // neighbor tile start (may be out of range)
        int j  = jt + n;                        // this lane's neighbor column (unclamped)
        int jc = j < 0 ? 0 : (j > NPTS - 1 ? NPTS - 1 : j);

        v2f bm;  // B matrix 4x16 f32, symmetric layout to A
        if (hi == 0) { bm.x = xb[jc * 3 + 0]; bm.y = xb[jc * 3 + 1]; }
        else         { bm.x = xb[jc * 3 + 2]; bm.y = 0.0f;           }

        v8f c = {};
        c = __builtin_amdgcn_wmma_f32_16x16x4_f32(
                /*neg_a=*/false, a, /*neg_b=*/false, bm,
                /*c_mod=*/(short)0, c, /*reuse_a=*/false, /*reuse_b=*/false);

        // C/D layout: VGPR v -> row M = v + 8*hi, this lane's col N = n.
        float sqc = sq[j - blockStart + 16];   // neighbor |x|^2 (garbage if clamped; masked later)
        #pragma unroll
        for (int v = 0; v < 8; ++v) {
            int M = v + 8 * hi;
            float sqr = sq[wave * 16 + 16 + M];               // row point |x|^2
            float d2  = sqr + sqc - 2.0f * c[v];
            float d   = __builtin_sqrtf(fmaxf(d2, 0.0f));
            drow[wave][M][tIdx * 16 + n] = d;
        }
    }
    __syncthreads();

    // ---- per-point gather + bitonic sort (32 elems, fully in registers) ----
    if (tid < PTS_PER_BLOCK) {
        const int m = tid & 15;          // point within its wave tile
        const int w = tid >> 4;          // owning wave's LDS region
        const int i = blockStart + w * 16 + m;   // global point index in batch
        const float INF = __builtin_inff();

        float v[32];
        #pragma unroll
        for (int s = 0; s < 32; ++s) {
            int off = (s < KNB) ? (s - KNB) : (s - KNB + 1);  // -16..-1, 1..16
            int col = m + 16 + off;                            // col in 48-wide row
            int j   = i + off;
            float d = drow[w][m][col];
            v[s] = (j >= 0 && j < NPTS) ? d : INF;
        }

        // bitonic sort network, ascending
        #pragma unroll
        for (int kk = 2; kk <= 32; kk <<= 1) {
            #pragma unroll
            for (int jj = kk >> 1; jj > 0; jj >>= 1) {
                #pragma unroll
                for (int t = 0; t < 32; ++t) {
                    int ixj = t ^ jj;
                    if (ixj > t) {
                        bool up = ((t & kk) == 0);
                        float lo = v[t], hh = v[ixj];
                        bool sw = up ? (lo > hh) : (lo < hh);
                        float nlo = sw ? hh : lo;
                        float nhi = sw ? lo : hh;
                        v[t] = nlo;
                        v[ixj] = nhi;
                    }
                }
            }
        }

        float* o = out + ((size_t)b * NPTS + i) * 32;
        #pragma unroll
        for (int s = 0; s < 32; ++s)
            o[s] = (v[s] == INF) ? 0.0f : v[s];
    }
}

extern "C" void kernel_launch(void* const* d_in, const int* in_sizes, int n_in,
                              void* d_out, int out_size, void* d_ws, size_t ws_size,
                              hipStream_t stream) {
    const float* x = (const float*)d_in[0];
    float* out = (float*)d_out;

    const int B = in_sizes[0] / (NPTS * 3);               // 4
    const int blocksPerBatch = NPTS / PTS_PER_BLOCK;      // 64
    dim3 grid(B * blocksPerBatch);
    dim3 block(256);
    LocalDists_band_wmma<<<grid, block, 0, stream>>>(x, out);
}